// SpatialNumDualDescriptorAB3_2740189135095
// MI455X (gfx1250) — compile-verified
//
#include <hip/hip_runtime.h>

typedef __attribute__((ext_vector_type(2))) float v2f;
typedef __attribute__((ext_vector_type(8))) float v8f;

namespace {
constexpr int kM = 16;                  // channels / matrix dim
constexpr int kS = 112;                 // input spatial extent
constexpr int kO = 111;                 // output spatial extent (VALID, window 2)
constexpr int kNW = kO * kO * kO;       // number of windows = 1,367,631
constexpr int kTiles = (kNW + 15) / 16; // 16 window-rows per wave
constexpr int kWavesPerBlock = 8;       // 256 threads (8 wave32)
constexpr int kThreads = kWavesPerBlock * 32;
}

__global__ __launch_bounds__(kThreads)
void window_avg_map16_wmma(const float* __restrict__ vec,
                           const float* __restrict__ Mmat,
                           float* __restrict__ out) {
  const int lane = threadIdx.x & 31;
  const int wave = threadIdx.x >> 5;
  const int tile = blockIdx.x * kWavesPerBlock + wave;
  if (tile >= kTiles) return;           // uniform per wave: EXEC stays all-1s

  const int half = lane >> 4;           // lane group (0/1)
  const int n    = lane & 15;           // A-row m / B,D column n
  const int coff = half << 1;           // K offset inside a K=4 slice

  const int wbase = tile << 4;

  // ---------- A: 2x2x2 window sums for row m = n ----------
  // Lane holds channels {4s + coff, 4s + coff + 1} for K-slice s (ISA 16x4 f32 A layout).
  int w = wbase + n;
  if (w >= kNW) w = kNW - 1;            // clamp loads only (tail tile); keeps EXEC full
  const int x = w % kO;
  const int t = w / kO;
  const int h = t % kO;
  const int d = t / kO;

  const float* base = vec + (((d * kS + h) * kS + x) * kM + coff);

  v2f a0 = {0.f, 0.f}, a1 = {0.f, 0.f}, a2 = {0.f, 0.f}, a3 = {0.f, 0.f};
#pragma unroll
  for (int dz = 0; dz < 2; ++dz) {
#pragma unroll
    for (int dy = 0; dy < 2; ++dy) {
#pragma unroll
      for (int dx = 0; dx < 2; ++dx) {
        const float* p = base + (dz * kS * kS + dy * kS + dx) * kM;
        a0 += *(const v2f*)(p + 0);     // 8B-aligned: channel stride is 16 floats
        a1 += *(const v2f*)(p + 4);
        a2 += *(const v2f*)(p + 8);
        a3 += *(const v2f*)(p + 12);
      }
    }
  }

  // ---------- B: B[k][n] = M[n][k] * (1/8)  (fold window mean into B) ----------
  const float* mp = Mmat + n * kM + coff;
  const v2f b0 = 0.125f * *(const v2f*)(mp + 0);
  const v2f b1 = 0.125f * *(const v2f*)(mp + 4);
  const v2f b2 = 0.125f * *(const v2f*)(mp + 8);
  const v2f b3 = 0.125f * *(const v2f*)(mp + 12);

  // ---------- D = A x B + C over K=16 via four K=4 WMMAs ----------
  v8f c = {0.f, 0.f, 0.f, 0.f, 0.f, 0.f, 0.f, 0.f};
  c = __builtin_amdgcn_wmma_f32_16x16x4_f32(false, a0, false, b0, (short)0, c, false, false);
  c = __builtin_amdgcn_wmma_f32_16x16x4_f32(false, a1, false, b1, (short)0, c, false, false);
  c = __builtin_amdgcn_wmma_f32_16x16x4_f32(false, a2, false, b2, (short)0, c, false, false);
  c = __builtin_amdgcn_wmma_f32_16x16x4_f32(false, a3, false, b3, (short)0, c, false, false);

  // ---------- store D: lane n holds rows m = r + 8*half ----------
  float* orow = out + (wbase + (half << 3)) * kM + n;
  if (tile + 1 < kTiles) {
    // Full tile (all but the single tail tile): unpredicated, clause-able stores.
    // Each store writes two contiguous 64B runs (lanes 0-15 row r, lanes 16-31 row r+8).
#pragma unroll
    for (int r = 0; r < 8; ++r)
      orow[r * kM] = c[r];
  } else {
    // Tail tile: kNW % 16 == 15, predicate per row.
#pragma unroll
    for (int r = 0; r < 8; ++r) {
      const int wout = wbase + (half << 3) + r;
      if (wout < kNW)
        orow[r * kM] = c[r];
    }
  }
}

extern "C" void kernel_launch(void* const* d_in, const int* in_sizes, int n_in,
                              void* d_out, int out_size, void* d_ws, size_t ws_size,
                              hipStream_t stream) {
  (void)in_sizes; (void)n_in; (void)out_size; (void)d_ws; (void)ws_size;
  const float* vec  = (const float*)d_in[0];   // (112,112,112,16) f32
  const float* Mmat = (const float*)d_in[1];   // (16,16) f32
  float* out = (float*)d_out;                  // (111^3, 16) f32

  const int blocks = (kTiles + kWavesPerBlock - 1) / kWavesPerBlock;
  window_avg_map16_wmma<<<blocks, kThreads, 0, stream>>>(vec, Mmat, out);
}